// BSPlines_90022514524376
// MI455X (gfx1250) — compile-verified
//
#include <hip/hip_runtime.h>

typedef float v4f __attribute__((ext_vector_type(4)));

#define TPB 256

// ---------------------------------------------------------------------------
// Kernel 1: symbolic de Boor. For the fixed FITPACK layout (n=5, k=3) the
// interval index m is only ever 3 or 4.  For fixed m, de Boor's recursion is
// a cubic polynomial in x; run the recursion on polynomial coefficients
// (alpha(x) = a0 + a1*x is linear, degrees grow 0->1->2->3) and emit the two
// coefficient sets + the breakpoint t[4] into workspace.
//   ws[0..3] = cubic coeffs (const,x,x^2,x^3) for m=3
//   ws[4..7] = cubic coeffs for m=4
//   ws[8]    = t[4]
// ---------------------------------------------------------------------------
__global__ void bspline_coeff_kernel(const float* __restrict__ t,
                                     const float* __restrict__ c,
                                     float* __restrict__ ws) {
  if (threadIdx.x != 0) return;
  float tt[9];
  for (int i = 0; i < 9; ++i) tt[i] = t[i];
  float cc[5];
  for (int i = 0; i < 5; ++i) cc[i] = c[i];

  for (int m = 3; m <= 4; ++m) {
    float d[4][4];  // d[j][p]: coefficient of x^p
    for (int j = 0; j < 4; ++j) {
      d[j][0] = cc[m - 3 + j];
      d[j][1] = 0.0f; d[j][2] = 0.0f; d[j][3] = 0.0f;
    }
    for (int r = 1; r <= 3; ++r) {
      for (int j = 3; j >= r; --j) {
        float left  = tt[j + m - 3];
        float right = tt[j + 1 + m - r];
        float denom = right - left;
        float inv   = (denom > 0.0f) ? (1.0f / denom) : 0.0f;
        float a0 = -left * inv;   // alpha(x) = a0 + a1*x  (0 when denom<=0)
        float a1 = inv;
        float diff[4], e[4];
        for (int i = 0; i < 4; ++i) diff[i] = d[j][i] - d[j - 1][i];
        for (int i = 0; i < 4; ++i) e[i] = d[j - 1][i] + a0 * diff[i];
        for (int i = 1; i < 4; ++i) e[i] += a1 * diff[i - 1];
        for (int i = 0; i < 4; ++i) d[j][i] = e[i];
      }
    }
    for (int i = 0; i < 4; ++i) ws[(m - 3) * 4 + i] = d[3][i];
  }
  ws[8] = tt[4];
}

// ---------------------------------------------------------------------------
// Per-element evaluation: pick interval (m=3 vs m=4), Horner, exact-zero mask.
// ~10 VALU-lane ops per 8 bytes of traffic -> memory bound at 23.3 TB/s.
// ---------------------------------------------------------------------------
__device__ __forceinline__ float eval_cubic(float x, float t4,
    float p0, float p1, float p2, float p3,
    float q0, float q1, float q2, float q3) {
  bool hi = (x >= t4);                 // m = hi ? 4 : 3 (matches clipped searchsorted)
  float a0 = hi ? q0 : p0;
  float a1 = hi ? q1 : p1;
  float a2 = hi ? q2 : p2;
  float a3 = hi ? q3 : p3;
  float r = fmaf(fmaf(fmaf(a3, x, a2), x, a1), x, a0);
  return (x == 0.0f) ? 0.0f : r;
}

// gfx1250 async copy global -> LDS (ASYNCcnt path).  The leading
// s_wait_dscnt 0 guarantees the ds_load that last consumed this LDS slot has
// fully completed before the async engine overwrites it (LDS ops may complete
// out of order with respect to async LDS writes).
__device__ __forceinline__ void async_load_b128(unsigned lds_addr, const v4f* g) {
  asm volatile("s_wait_dscnt 0x0\n\t"
               "global_load_async_to_lds_b128 %0, %1, off"
               :: "v"(lds_addr), "v"(g)
               : "memory");
}

__global__ __launch_bounds__(TPB) void bspline_eval_kernel(
    const float* __restrict__ in, const float* __restrict__ ws,
    float* __restrict__ out, int n, int n4) {
  __shared__ v4f buf[2][TPB];   // 8 KB double buffer (one b128 slot per lane)

  // Uniform spline parameters (9 floats, L2-resident after kernel 1).
  const float p0 = ws[0], p1 = ws[1], p2 = ws[2], p3 = ws[3];
  const float q0 = ws[4], q1 = ws[5], q2 = ws[6], q3 = ws[7];
  const float t4 = ws[8];

  const int tid    = (int)threadIdx.x;
  const int stride = (int)gridDim.x * TPB;
  const v4f* in4   = (const v4f*)in;
  v4f*       out4  = (v4f*)out;

  int idx = (int)blockIdx.x * TPB + tid;

  // Full (unguarded) iterations: every thread in range, uniform trip count.
  const int T = n4 / stride;

  unsigned lds_addr[2];
  lds_addr[0] = (unsigned)(unsigned long long)(&buf[0][tid]);  // low 32 bits = LDS offset
  lds_addr[1] = (unsigned)(unsigned long long)(&buf[1][tid]);

  if (T > 0) async_load_b128(lds_addr[0], in4 + idx);

  for (int k = 0; k < T; ++k) {
    int next = idx + stride;
    if (k + 1 < T) {
      async_load_b128(lds_addr[(k + 1) & 1], in4 + next);   // depth-2 pipeline
      asm volatile("s_wait_asynccnt 0x1" ::: "memory");     // stage k landed
    } else {
      asm volatile("s_wait_asynccnt 0x0" ::: "memory");     // drain
    }
    v4f x = buf[k & 1][tid];                                // ds_load_b128
    v4f y;
    y.x = eval_cubic(x.x, t4, p0, p1, p2, p3, q0, q1, q2, q3);
    y.y = eval_cubic(x.y, t4, p0, p1, p2, p3, q0, q1, q2, q3);
    y.z = eval_cubic(x.z, t4, p0, p1, p2, p3, q0, q1, q2, q3);
    y.w = eval_cubic(x.w, t4, p0, p1, p2, p3, q0, q1, q2, q3);
    __builtin_nontemporal_store(y, out4 + idx);             // NT: keep input in L2
    idx = next;
  }

  // float4 tail (idx == base + T*stride here; remainder < stride).
  if (idx < n4) {
    v4f x = in4[idx];
    v4f y;
    y.x = eval_cubic(x.x, t4, p0, p1, p2, p3, q0, q1, q2, q3);
    y.y = eval_cubic(x.y, t4, p0, p1, p2, p3, q0, q1, q2, q3);
    y.z = eval_cubic(x.z, t4, p0, p1, p2, p3, q0, q1, q2, q3);
    y.w = eval_cubic(x.w, t4, p0, p1, p2, p3, q0, q1, q2, q3);
    __builtin_nontemporal_store(y, out4 + idx);
  }

  // scalar tail (n % 4, at most 3 elements globally).
  int s = (n4 << 2) + (int)blockIdx.x * TPB + tid;
  if (s < n) {
    float x = in[s];
    out[s] = eval_cubic(x, t4, p0, p1, p2, p3, q0, q1, q2, q3);
  }
}

extern "C" void kernel_launch(void* const* d_in, const int* in_sizes, int n_in,
                              void* d_out, int out_size, void* d_ws, size_t ws_size,
                              hipStream_t stream) {
  const float* imgs = (const float*)d_in[0];  // 64*3*512*512 fp32
  const float* t    = (const float*)d_in[1];  // 9 fp32
  const float* c    = (const float*)d_in[2];  // 5 fp32
  float* out = (float*)d_out;
  float* ws  = (float*)d_ws;

  int n  = in_sizes[0];
  int n4 = n >> 2;

  // 1) derive the two cubic polynomials from (t, c)
  bspline_coeff_kernel<<<1, 32, 0, stream>>>(t, c, ws);

  // 2) stream imgs through the async-LDS pipeline
  int blocks = 2048;                       // 524288 threads -> 24 full iters here
  int need = (n4 + TPB - 1) / TPB;
  if (need < 1) need = 1;
  if (blocks > need) blocks = need;
  bspline_eval_kernel<<<blocks, TPB, 0, stream>>>(imgs, ws, out, n, n4);
}